// Graph_Attention_new_41815801593869
// MI455X (gfx1250) — compile-verified
//
#include <hip/hip_runtime.h>
#include <hip/hip_bf16.h>

// ---------------------------------------------------------------------------
// Graph attention block for MI455X (gfx1250, wave32).
// All matrix math through V_WMMA_F32_16X16X4_F32 (fp32 in/out, matches ref).
// Flash-attention fusion: the [B,4096,4096] score matrix never hits HBM.
// ---------------------------------------------------------------------------

typedef float v2f __attribute__((ext_vector_type(2)));
typedef float v4f __attribute__((ext_vector_type(4)));
typedef float v8f __attribute__((ext_vector_type(8)));

#define WMMA4(a, b, c) \
  __builtin_amdgcn_wmma_f32_16x16x4_f32(false, (a), false, (b), (short)0, (c), false, false)

// ---- problem constants ----
static constexpr int B_  = 4;
static constexpr int H_  = 128;
static constexpr int W_  = 128;
static constexpr int C_  = 64;
static constexpr int h_  = 64;      // H/2
static constexpr int w_  = 64;      // W/2
static constexpr int Nx_ = h_ * w_; // 4096 tokens per batch
static constexpr int NP_ = B_ * Nx_; // 16384 downsampled pixels total

// ---- workspace layout (float offsets). total ~38 MB ----
static constexpr size_t XT  = 0;                    // xf_trans  [NP,64]
static constexpr size_t ZT  = XT  + (size_t)NP_ * C_; // zf_trans
static constexpr size_t XG  = ZT  + (size_t)NP_ * C_; // conv(xf,W_g)+b_g (pre-BN)
static constexpr size_t ZG  = XG  + (size_t)NP_ * C_; // conv(zf,W_g)+b_g (pre-BN)
static constexpr size_t EMB = ZG  + (size_t)NP_ * C_; // attention output [NP,64]
static constexpr size_t Y_  = EMB + (size_t)NP_ * C_; // deconv pre-BN [B,128,128,64]
static constexpr size_t ST  = Y_  + (size_t)B_ * H_ * W_ * C_; // stats block (768 f)
// stats block layout (relative to ST):
//   +0 sum_xg[64]  +64 sq_xg[64]  +128 sum_zg[64]  +192 sq_zg[64]
//   +256 scale_x[64] +320 shift_x[64] +384 scale_z[64] +448 shift_z[64]
//   +512 sum_y[64] +576 sq_y[64] +640 scale_y[64] +704 shift_y[64]

// ---- wave32 helpers: reductions within each 16-lane half ----
static __device__ __forceinline__ float redmax16(float v) {
  v = fmaxf(v, __shfl_xor(v, 1, 32));
  v = fmaxf(v, __shfl_xor(v, 2, 32));
  v = fmaxf(v, __shfl_xor(v, 4, 32));
  v = fmaxf(v, __shfl_xor(v, 8, 32));
  return v;
}
static __device__ __forceinline__ float redsum16(float v) {
  v += __shfl_xor(v, 1, 32);
  v += __shfl_xor(v, 2, 32);
  v += __shfl_xor(v, 4, 32);
  v += __shfl_xor(v, 8, 32);
  return v;
}

// Per-wave LDS producer->consumer fence using the CDNA5 split DS counter.
// Waves run in lockstep on one SIMD32; s_wait_dscnt 0 guarantees all DS ops
// of this wave (all lanes) have completed.
static __device__ __forceinline__ void wave_lds_fence() {
  __builtin_amdgcn_wave_barrier();
  asm volatile("s_wait_dscnt 0" ::: "memory");
  __builtin_amdgcn_wave_barrier();
}

// ---------------------------------------------------------------------------
// 0) zero the stat accumulators (harness poisons ws with 0xAA)
// ---------------------------------------------------------------------------
__global__ __launch_bounds__(256) void k_init(float* wsf) {
  for (int i = threadIdx.x; i < 768; i += 256) wsf[ST + i] = 0.0f;
}

// ---------------------------------------------------------------------------
// 1) fused projection convs: 4 GEMMs [16384 x 256] @ [256 x 64].
//    Each wave owns 16 output pixels x all 64 channels for all 4 outputs,
//    sharing the im2col A-fragment between the two convs on each input.
// ---------------------------------------------------------------------------
__global__ __launch_bounds__(128) void k_conv_proj(
    const float* __restrict__ xf, const float* __restrict__ zf,
    const float* __restrict__ Wq, const float* __restrict__ bq,
    const float* __restrict__ Wss, const float* __restrict__ bs,
    const float* __restrict__ Wg, const float* __restrict__ bg,
    float* __restrict__ wsf) {
  const int lane = threadIdx.x & 31;
  const int wv   = threadIdx.x >> 5;
  const int hi   = lane >> 4;
  const int l16  = lane & 15;

  const int pbase = blockIdx.x * 64 + wv * 16;   // 16 pixels per wave
  const int prow  = pbase + l16;                 // A-matrix row for this lane
  const int b  = prow >> 12;
  const int rr = prow & 4095;
  const int oh = rr >> 6;
  const int ow = rr & 63;
  // patch base: x[b, 2*oh + kh, 2*ow .. 2*ow+1, :] is contiguous 128 floats
  const size_t pat = (((size_t)b * H_ + 2 * oh) * W_ + 2 * ow) * C_;
  const float* xr = xf + pat;
  const float* zr = zf + pat;

  v8f aq[4], ag[4], as_[4], azg[4];
#pragma unroll
  for (int nt = 0; nt < 4; ++nt) {
    aq[nt] = v8f{0, 0, 0, 0, 0, 0, 0, 0};
    ag[nt] = v8f{0, 0, 0, 0, 0, 0, 0, 0};
    as_[nt] = v8f{0, 0, 0, 0, 0, 0, 0, 0};
    azg[nt] = v8f{0, 0, 0, 0, 0, 0, 0, 0};
  }

  for (int k = 0; k < 64; ++k) {           // K = 256, 4 per WMMA
    const int kidx = 4 * k + 2 * hi;       // this lane's first K column
    const int kh   = kidx >> 7;            // kernel row
    const int r128 = kidx & 127;           // (kw,ci) within the contiguous run
    const v2f ax = *(const v2f*)(xr + (size_t)kh * (W_ * C_) + r128);
    const v2f az = *(const v2f*)(zr + (size_t)kh * (W_ * C_) + r128);
#pragma unroll
    for (int nt = 0; nt < 4; ++nt) {
      const int col = nt * 16 + l16;
      v2f wqf, wsf2, wgf;
      wqf.x = Wq[(size_t)kidx * 64 + col];
      wqf.y = Wq[(size_t)(kidx + 1) * 64 + col];
      wsf2.x = Wss[(size_t)kidx * 64 + col];
      wsf2.y = Wss[(size_t)(kidx + 1) * 64 + col];
      wgf.x = Wg[(size_t)kidx * 64 + col];
      wgf.y = Wg[(size_t)(kidx + 1) * 64 + col];
      aq[nt]  = WMMA4(ax, wqf, aq[nt]);
      ag[nt]  = WMMA4(ax, wgf, ag[nt]);
      as_[nt] = WMMA4(az, wsf2, as_[nt]);
      azg[nt] = WMMA4(az, wgf, azg[nt]);
    }
  }

  // epilogue: bias, store, accumulate BN stats for the two g-conv outputs
#pragma unroll
  for (int nt = 0; nt < 4; ++nt) {
    const int col = nt * 16 + l16;
    const float vbq = bq[col], vbs = bs[col], vbg = bg[col];
    float sx = 0.f, qx = 0.f, sz = 0.f, qz = 0.f;
#pragma unroll
    for (int i = 0; i < 8; ++i) {
      const int row = pbase + i + 8 * hi;
      const size_t o64 = (size_t)row * 64 + col;
      wsf[XT + o64] = aq[nt][i] + vbq;
      wsf[ZT + o64] = as_[nt][i] + vbs;
      const float gx = ag[nt][i] + vbg;
      const float gz = azg[nt][i] + vbg;
      wsf[XG + o64] = gx;
      wsf[ZG + o64] = gz;
      sx += gx; qx += gx * gx;
      sz += gz; qz += gz * gz;
    }
    sx += __shfl_xor(sx, 16, 32); qx += __shfl_xor(qx, 16, 32);
    sz += __shfl_xor(sz, 16, 32); qz += __shfl_xor(qz, 16, 32);
    if (lane < 16) {
      atomicAdd(&wsf[ST + 0 + col], sx);
      atomicAdd(&wsf[ST + 64 + col], qx);
      atomicAdd(&wsf[ST + 128 + col], sz);
      atomicAdd(&wsf[ST + 192 + col], qz);
    }
  }
}

// ---------------------------------------------------------------------------
// 2) finalize g-BN: scale = gamma*rsqrt(var+eps), shift = beta - mu*scale
// ---------------------------------------------------------------------------
__global__ void k_fin_g(float* wsf, const float* __restrict__ gamma,
                        const float* __restrict__ beta) {
  const int t = threadIdx.x;
  if (t >= 128) return;
  const int ten = t >> 6, c = t & 63;
  const float n = 16384.0f;
  const float mu  = wsf[ST + ten * 128 + c] / n;
  const float var = wsf[ST + 64 + ten * 128 + c] / n - mu * mu;
  const float sc  = gamma[c] * rsqrtf(var + 1e-3f);
  wsf[ST + 256 + ten * 128 + c] = sc;
  wsf[ST + 320 + ten * 128 + c] = beta[c] - mu * sc;
}

// ---------------------------------------------------------------------------
// 3) flash attention. Per wave: 16 queries x 4096 keys (16-key tiles).
//    logits (16 WMMA) -> online softmax (xor-shuffle row reductions)
//    -> P through per-wave LDS tile (D-layout -> A-layout transpose)
//    -> P @ V (16 WMMA), V = relu(bn(zg_pre)) computed on the fly.
// ---------------------------------------------------------------------------
__global__ __launch_bounds__(128) void k_attn(float* __restrict__ wsf) {
  __shared__ float pbuf[4 * 16 * 16];  // per-wave 16x16 P tile
  const int lane = threadIdx.x & 31;
  const int wv   = threadIdx.x >> 5;
  const int hi   = lane >> 4;
  const int l16  = lane & 15;
  float* pt = pbuf + wv * 256;

  const int b     = blockIdx.x >> 6;          // 64 blocks per batch
  const int qblk  = blockIdx.x & 63;
  const int qbase = qblk * 64 + wv * 16;

  const float* Xt = wsf + XT + (size_t)b * Nx_ * C_;
  const float* Zt = wsf + ZT + (size_t)b * Nx_ * C_;
  const float* Zg = wsf + ZG + (size_t)b * Nx_ * C_;

  // preload Q fragments for all 16 K-steps (K=64)
  v2f qa[16];
  const int qrow = qbase + l16;
#pragma unroll
  for (int k = 0; k < 16; ++k)
    qa[k] = *(const v2f*)(Xt + (size_t)qrow * 64 + 4 * k + 2 * hi);

  // per-lane value-channel BN params (channel = nt*16 + l16)
  float scz[4], shz[4];
#pragma unroll
  for (int nt = 0; nt < 4; ++nt) {
    const int c = nt * 16 + l16;
    scz[nt] = wsf[ST + 384 + c];
    shz[nt] = wsf[ST + 448 + c];
  }

  v8f o[4];
  float m[8], lsum[8];
#pragma unroll
  for (int nt = 0; nt < 4; ++nt) o[nt] = v8f{0, 0, 0, 0, 0, 0, 0, 0};
#pragma unroll
  for (int i = 0; i < 8; ++i) { m[i] = -3.0e38f; lsum[i] = 0.0f; }

  for (int t = 0; t < Nx_ / 16; ++t) {
    const int n0 = t * 16;

    // --- logits S = Q K^T for this key tile ---
    v8f s = v8f{0, 0, 0, 0, 0, 0, 0, 0};
    const int krow = n0 + l16;
#pragma unroll
    for (int k = 0; k < 16; ++k) {
      const v2f kb = *(const v2f*)(Zt + (size_t)krow * 64 + 4 * k + 2 * hi);
      s = WMMA4(qa[k], kb, s);
    }

    // --- online softmax over this tile ---
    float pv[8];
#pragma unroll
    for (int i = 0; i < 8; ++i) {
      const float sv = s[i];
      const float mt = redmax16(sv);
      const float mn = fmaxf(m[i], mt);
      const float f  = __expf(m[i] - mn);
      const float p  = __expf(sv - mn);
      lsum[i] = lsum[i] * f + redsum16(p);
      m[i] = mn;
      pv[i] = p;
#pragma unroll
      for (int nt = 0; nt < 4; ++nt) o[nt][i] *= f;
    }

    // --- P (D-layout) -> LDS -> A-layout fragments ---
#pragma unroll
    for (int i = 0; i < 8; ++i) pt[(i + 8 * hi) * 16 + l16] = pv[i];
    wave_lds_fence();
    v2f pa[4];
#pragma unroll
    for (int k = 0; k < 4; ++k)
      pa[k] = *(const v2f*)(pt + l16 * 16 + 4 * k + 2 * hi);
    wave_lds_fence();  // WAR: loads done before next tile's stores

    // --- O += P @ V,  V = relu(scale*zg_pre + shift) ---
#pragma unroll
    for (int k = 0; k < 4; ++k) {
      const int key = n0 + 4 * k + 2 * hi;
#pragma unroll
      for (int nt = 0; nt < 4; ++nt) {
        const int c = nt * 16 + l16;
        v2f vb;
        vb.x = fmaxf(Zg[(size_t)key * 64 + c] * scz[nt] + shz[nt], 0.0f);
        vb.y = fmaxf(Zg[(size_t)(key + 1) * 64 + c] * scz[nt] + shz[nt], 0.0f);
        o[nt] = WMMA4(pa[k], vb, o[nt]);
      }
    }
  }

  // --- normalize and store embedding ---
  float inv[8];
#pragma unroll
  for (int i = 0; i < 8; ++i) inv[i] = 1.0f / lsum[i];
#pragma unroll
  for (int nt = 0; nt < 4; ++nt) {
    const int c = nt * 16 + l16;
#pragma unroll
    for (int i = 0; i < 8; ++i) {
      const int q = qbase + i + 8 * hi;
      wsf[EMB + ((size_t)b * Nx_ + q) * 64 + c] = o[nt][i] * inv[i];
    }
  }
}

// ---------------------------------------------------------------------------
// 4) transposed conv as GEMM: M=16384 pixels, K=128 (concat channels),
//    N=(kh,kw,co). Each of 4 waves owns one (kh,kw) quadrant of the same
//    16 pixels. Writes pre-BN y and accumulates fi-BN stats.
// ---------------------------------------------------------------------------
__global__ __launch_bounds__(128) void k_deconv(
    float* __restrict__ wsf, const float* __restrict__ Wfi,
    const float* __restrict__ bfi) {
  const int lane = threadIdx.x & 31;
  const int wv   = threadIdx.x >> 5;
  const int hi   = lane >> 4;
  const int l16  = lane & 15;
  const int kh = wv >> 1, kw = wv & 1;
  const int q128 = (kh * 2 + kw) * 128;

  const int pbase = blockIdx.x * 16;
  const int prow  = pbase + l16;
  const float* Emb = wsf + EMB;
  const float* Xg  = wsf + XG;

  v8f acc[4];
#pragma unroll
  for (int nt = 0; nt < 4; ++nt) acc[nt] = v8f{0, 0, 0, 0, 0, 0, 0, 0};

  for (int k = 0; k < 32; ++k) {          // K = 128
    const int ci0 = 4 * k + 2 * hi;
    v2f a;
    if (k < 16) {                          // first 64 channels: embedding
      a = *(const v2f*)(Emb + (size_t)prow * 64 + ci0);
    } else {                               // last 64: relu(bn(xg_pre))
      const int c = ci0 - 64;
      const v2f x  = *(const v2f*)(Xg + (size_t)prow * 64 + c);
      const v2f sc = *(const v2f*)(wsf + ST + 256 + c);
      const v2f sh = *(const v2f*)(wsf + ST + 320 + c);
      a.x = fmaxf(x.x * sc.x + sh.x, 0.0f);
      a.y = fmaxf(x.y * sc.y + sh.y, 0.0f);
    }
#pragma unroll
    for (int nt = 0; nt < 4; ++nt) {
      const int co = nt * 16 + l16;
      v2f wb;
      wb.x = Wfi[(size_t)(q128 + ci0) * 64 + co];
      wb.y = Wfi[(size_t)(q128 + ci0 + 1) * 64 + co];
      acc[nt] = WMMA4(a, wb, acc[nt]);
    }
  }

#pragma unroll
  for (int nt = 0; nt < 4; ++nt) {
    const int co = nt * 16 + l16;
    const float bi = bfi[co];
    float s = 0.f, q = 0.f;
#pragma unroll
    for (int i = 0; i < 8; ++i) {
      const int p  = pbase + i + 8 * hi;
      const int b  = p >> 12;
      const int rr = p & 4095;
      const int oh = rr >> 6, ow = rr & 63;
      const float v = acc[nt][i] + bi;
      wsf[Y_ + (((size_t)b * H_ + 2 * oh + kh) * W_ + 2 * ow + kw) * 64 + co] = v;
      s += v; q += v * v;
    }
    s += __shfl_xor(s, 16, 32);
    q += __shfl_xor(q, 16, 32);
    if (lane < 16) {
      atomicAdd(&wsf[ST + 512 + co], s);
      atomicAdd(&wsf[ST + 576 + co], q);
    }
  }
}

// ---------------------------------------------------------------------------
// 5) finalize fi-BN
// ---------------------------------------------------------------------------
__global__ void k_fin_fi(float* wsf, const float* __restrict__ gamma,
                         const float* __restrict__ beta) {
  const int c = threadIdx.x;
  if (c >= 64) return;
  const float n = (float)(B_ * H_ * W_);  // 65536
  const float mu  = wsf[ST + 512 + c] / n;
  const float var = wsf[ST + 576 + c] / n - mu * mu;
  const float sc  = gamma[c] * rsqrtf(var + 1e-3f);
  wsf[ST + 640 + c] = sc;
  wsf[ST + 704 + c] = beta[c] - mu * sc;
}

// ---------------------------------------------------------------------------
// 6) out = relu(scale*y + shift), float4 traffic (channel dim = 64)
// ---------------------------------------------------------------------------
__global__ __launch_bounds__(256) void k_apply(const float* __restrict__ wsf,
                                               float* __restrict__ out) {
  const int i4 = blockIdx.x * 256 + threadIdx.x;   // 1048576 float4s
  const v4f y  = *(const v4f*)(wsf + Y_ + (size_t)i4 * 4);
  const int cg = (i4 & 15) * 4;                    // channel group
  const v4f sc = *(const v4f*)(wsf + ST + 640 + cg);
  const v4f sh = *(const v4f*)(wsf + ST + 704 + cg);
  v4f r;
  r.x = fmaxf(y.x * sc.x + sh.x, 0.0f);
  r.y = fmaxf(y.y * sc.y + sh.y, 0.0f);
  r.z = fmaxf(y.z * sc.z + sh.z, 0.0f);
  r.w = fmaxf(y.w * sc.w + sh.w, 0.0f);
  *(v4f*)(out + (size_t)i4 * 4) = r;
}

// ---------------------------------------------------------------------------
extern "C" void kernel_launch(void* const* d_in, const int* in_sizes, int n_in,
                              void* d_out, int out_size, void* d_ws,
                              size_t ws_size, hipStream_t stream) {
  (void)in_sizes; (void)n_in; (void)out_size; (void)ws_size;
  const float* zf  = (const float*)d_in[0];
  const float* xf  = (const float*)d_in[1];
  const float* Wq  = (const float*)d_in[2];
  const float* bq  = (const float*)d_in[3];
  const float* Wss = (const float*)d_in[4];
  const float* bs  = (const float*)d_in[5];
  const float* Wg  = (const float*)d_in[6];
  const float* bg  = (const float*)d_in[7];
  const float* gga = (const float*)d_in[8];
  const float* gbe = (const float*)d_in[9];
  const float* Wfi = (const float*)d_in[10];
  const float* bfi = (const float*)d_in[11];
  const float* fga = (const float*)d_in[12];
  const float* fbe = (const float*)d_in[13];
  float* wsf = (float*)d_ws;
  float* out = (float*)d_out;

  k_init<<<1, 256, 0, stream>>>(wsf);
  k_conv_proj<<<NP_ / 64, 128, 0, stream>>>(xf, zf, Wq, bq, Wss, bs, Wg, bg, wsf);
  k_fin_g<<<1, 128, 0, stream>>>(wsf, gga, gbe);
  k_attn<<<B_ * (Nx_ / 64), 128, 0, stream>>>(wsf);
  k_deconv<<<NP_ / 16, 128, 0, stream>>>(wsf, Wfi, bfi);
  k_fin_fi<<<1, 64, 0, stream>>>(wsf, fga, fbe);
  k_apply<<<(B_ * H_ * W_ * C_ / 4) / 256, 256, 0, stream>>>(wsf, out);
}